// PillarMaxPoolingV1a_51015621542404
// MI455X (gfx1250) — compile-verified
//
#include <hip/hip_runtime.h>
#include <hip/hip_bf16.h>

typedef __attribute__((ext_vector_type(2))) float v2f;
typedef __attribute__((ext_vector_type(8))) float v8f;

#define N_PTS   1200000
#define N_PIL   60000
#define BB      4
#define HH      400
#define WW      400
#define NBLK    9375          // N_PTS / 128
#define TBL_SZ  (BB*HH*WW)    // 640000
#define OUT_SZ  (N_PIL*64)    // 3840000

__device__ __forceinline__ v8f vz8() {
    v8f z;
#pragma unroll
    for (int i = 0; i < 8; i++) z[i] = 0.0f;
    return z;
}

// ---------------- init: table = -1, out = 0 ----------------
__global__ __launch_bounds__(256) void k_init(int* __restrict__ table, float* __restrict__ out) {
    int i = blockIdx.x * 256 + threadIdx.x;
    if (i < TBL_SZ) table[i] = -1;
    if (i < OUT_SZ) out[i] = 0.0f;
}

// ---------------- build pillar hash table (last/max index wins) ----------------
__global__ __launch_bounds__(256) void k_table(const int* __restrict__ pind, int* __restrict__ table) {
    int i = blockIdx.x * 256 + threadIdx.x;
    if (i < N_PIL) {
        int b = pind[3*i], y = pind[3*i+1], x = pind[3*i+2];
        atomicMax(&table[(b*HH + y)*WW + x], i);
    }
}

// ---------------- BN parameter computation (deterministic tree reduce) ----------------
__global__ __launch_bounds__(256) void k_bnparam(const float* __restrict__ psum, const float* __restrict__ psq,
                                                 float* __restrict__ scale, float* __restrict__ shift,
                                                 const float* __restrict__ gamma, const float* __restrict__ beta) {
    __shared__ float ss[256], sq[256];
    int c = blockIdx.x, tid = threadIdx.x;
    float s = 0.0f, q = 0.0f;
    for (int i = tid; i < NBLK; i += 256) { s += psum[c*NBLK + i]; q += psq[c*NBLK + i]; }
    ss[tid] = s; sq[tid] = q; __syncthreads();
    for (int o = 128; o > 0; o >>= 1) {
        if (tid < o) { ss[tid] += ss[tid+o]; sq[tid] += sq[tid+o]; }
        __syncthreads();
    }
    if (tid == 0) {
        float mean = ss[0] / (float)N_PTS;
        float var  = sq[0] / (float)N_PTS - mean*mean;
        float sc   = gamma[c] / sqrtf(var + 1e-3f);
        scale[c] = sc;
        shift[c] = beta[c] - mean*sc;
    }
}

// ---------------- fused point-MLP passes ----------------
// PASS 1: GEMM1, emit per-block channel sum/sumsq of h1
// PASS 2: GEMM1 + BN1 + ReLU + GEMM2, emit sum/sumsq of h2
// PASS 3: full MLP + BN2 + ReLU + scatter-max to pillars
template<int PASS>
__global__ __launch_bounds__(256) void k_layer(
    const float* __restrict__ xyz, const int* __restrict__ pbidx,
    const float* __restrict__ pfeat, const int* __restrict__ table,
    const float* __restrict__ w1g, const float* __restrict__ w2g,
    const float* __restrict__ bng, float* __restrict__ psum,
    float* __restrict__ psq, float* __restrict__ out)
{
    constexpr int GS = 37;   // g tile row stride (conflict-free A loads)
    constexpr int WS = 72;   // weight tile row stride (conflict-free B loads)
    constexpr int AS = 69;   // h1 tile row stride (conflict-free A loads)
    constexpr int B1OFF = 128*GS;                                   // 4736
    constexpr int UEND  = (PASS == 1) ? (B1OFF + 36*WS) : (128*AS); // 7328 : 8832
    constexpr int W2OFF = UEND;
    constexpr int PSOFF = W2OFF + ((PASS >= 2) ? 64*WS : 0);
    constexpr int PQOFF = PSOFF + 1024;
    constexpr int BNOFF = PQOFF + 1024;
    constexpr int SIOFF = BNOFF + ((PASS >= 2) ? 256 : 0);
    constexpr int TOTAL = SIOFF + ((PASS == 3) ? 128 : 0);
    __shared__ float sm[TOTAL];

    const int tid  = threadIdx.x;
    const int wave = tid >> 5;
    const int lane = tid & 31;
    const int l    = lane & 15;
    const int half = lane >> 4;
    const int blk  = blockIdx.x;
    const int p0   = blk * 128;

    // ---- stage point features (coalesced float4) into g tile ----
    const float4* pf4 = (const float4*)pfeat;
    for (int i = tid; i < 128*8; i += 256) {
        int p = i >> 3, q = i & 7;
        float4 v = pf4[(size_t)(p0 + p) * 8 + q];
        float* dst = &sm[p*GS + q*4];
        dst[0] = v.x; dst[1] = v.y; dst[2] = v.z; dst[3] = v.w;
    }
    // ---- per-point geometry: rel features (+ pillar lookup on pass 3) ----
    if (tid < 128) {
        int p = p0 + tid;
        float x = xyz[3*p], y = xyz[3*p+1], z = xyz[3*p+2];
        int xi = (int)floorf((x - (-80.0f)) / 0.4f); xi = min(max(xi, 0), WW-1);
        int yi = (int)floorf((y - (-80.0f)) / 0.4f); yi = min(max(yi, 0), HH-1);
        float xc = -80.0f + ((float)xi + 0.5f) * 0.4f;
        float yc = -80.0f + ((float)yi + 0.5f) * 0.4f;
        sm[tid*GS + 32] = x - xc;
        sm[tid*GS + 33] = y - yc;
        sm[tid*GS + 34] = z - (-1.0f);
        sm[tid*GS + 35] = 0.0f;     // K padding
        if (PASS == 3) {
            int cell = (pbidx[p]*HH + yi)*WW + xi;
            ((int*)sm)[SIOFF + tid] = table[cell];
        }
    }
    // ---- stage w1 (transposed to [k][n]) ----
    for (int i = tid; i < 64*35; i += 256) {
        int n = i / 35, k = i - n*35;
        sm[B1OFF + k*WS + n] = w1g[i];
    }
    if (tid < 64) sm[B1OFF + 35*WS + tid] = 0.0f;   // padded K row
    if (PASS >= 2) {
        for (int i = tid; i < 64*64; i += 256) {
            int n = i >> 6, k = i & 63;
            sm[W2OFF + k*WS + n] = w2g[i];
        }
        if (tid < ((PASS == 3) ? 256 : 128)) sm[BNOFF + tid] = bng[tid];
    }
    __syncthreads();

    // ---- GEMM1: 16x64 tile per wave, fp32 WMMA, K = 36 ----
    const int rowA = wave*16 + l;
    v8f acc[4];
#pragma unroll
    for (int t = 0; t < 4; t++) acc[t] = vz8();
#pragma unroll
    for (int k0 = 0; k0 < 36; k0 += 4) {
        v2f a;
        a[0] = sm[rowA*GS + k0 + half*2];
        a[1] = sm[rowA*GS + k0 + half*2 + 1];
#pragma unroll
        for (int t = 0; t < 4; t++) {
            v2f b;
            b[0] = sm[B1OFF + (k0 + half*2)*WS + t*16 + l];
            b[1] = sm[B1OFF + (k0 + half*2 + 1)*WS + t*16 + l];
            acc[t] = __builtin_amdgcn_wmma_f32_16x16x4_f32(false, a, false, b, (short)0, acc[t], false, false);
        }
    }

    if (PASS == 1) {
        // per-lane partial sums of raw h1, fixed LDS slots (deterministic)
#pragma unroll
        for (int t = 0; t < 4; t++) {
            float s = 0.0f, q = 0.0f;
#pragma unroll
            for (int r = 0; r < 8; r++) { float v = acc[t][r]; s += v; q += v*v; }
            int c = t*16 + l;
            sm[PSOFF + c*16 + wave*2 + half] = s;
            sm[PQOFF + c*16 + wave*2 + half] = q;
        }
        __syncthreads();
        if (tid < 64) {
            float s = 0.0f, q = 0.0f;
#pragma unroll
            for (int j = 0; j < 16; j++) { s += sm[PSOFF + tid*16 + j]; q += sm[PQOFF + tid*16 + j]; }
            psum[tid*NBLK + blk] = s;
            psq [tid*NBLK + blk] = q;
        }
        return;
    }

    if (PASS >= 2) {
        __syncthreads();   // g/w1 region dead; safe to overlay with h1 tile
        // BN1 + ReLU, write normalized h1 into A-tile layout
#pragma unroll
        for (int t = 0; t < 4; t++) {
            int c = t*16 + l;
            float sc = sm[BNOFF + c], sh = sm[BNOFF + 64 + c];
#pragma unroll
            for (int r = 0; r < 8; r++) {
                float v = fmaxf(fmaf(acc[t][r], sc, sh), 0.0f);
                int row = wave*16 + half*8 + r;
                sm[row*AS + t*16 + l] = v;
            }
        }
        __syncthreads();

        // ---- GEMM2: K = 64 ----
        v8f acc2[4];
#pragma unroll
        for (int t = 0; t < 4; t++) acc2[t] = vz8();
#pragma unroll
        for (int k0 = 0; k0 < 64; k0 += 4) {
            v2f a;
            a[0] = sm[rowA*AS + k0 + half*2];
            a[1] = sm[rowA*AS + k0 + half*2 + 1];
#pragma unroll
            for (int t = 0; t < 4; t++) {
                v2f b;
                b[0] = sm[W2OFF + (k0 + half*2)*WS + t*16 + l];
                b[1] = sm[W2OFF + (k0 + half*2 + 1)*WS + t*16 + l];
                acc2[t] = __builtin_amdgcn_wmma_f32_16x16x4_f32(false, a, false, b, (short)0, acc2[t], false, false);
            }
        }

        if (PASS == 2) {
#pragma unroll
            for (int t = 0; t < 4; t++) {
                float s = 0.0f, q = 0.0f;
#pragma unroll
                for (int r = 0; r < 8; r++) { float v = acc2[t][r]; s += v; q += v*v; }
                int c = t*16 + l;
                sm[PSOFF + c*16 + wave*2 + half] = s;
                sm[PQOFF + c*16 + wave*2 + half] = q;
            }
            __syncthreads();
            if (tid < 64) {
                float s = 0.0f, q = 0.0f;
#pragma unroll
                for (int j = 0; j < 16; j++) { s += sm[PSOFF + tid*16 + j]; q += sm[PQOFF + tid*16 + j]; }
                psum[tid*NBLK + blk] = s;
                psq [tid*NBLK + blk] = q;
            }
            return;
        }

        // PASS 3: BN2 + ReLU + scatter-max (int atomicMax on non-negative float bits)
#pragma unroll
        for (int t = 0; t < 4; t++) {
            int c = t*16 + l;
            float sc = sm[BNOFF + 128 + c], sh = sm[BNOFF + 192 + c];
#pragma unroll
            for (int r = 0; r < 8; r++) {
                float v = fmaxf(fmaf(acc2[t][r], sc, sh), 0.0f);
                int row = wave*16 + half*8 + r;
                int si = ((const int*)sm)[SIOFF + row];
                if (si >= 0) atomicMax((int*)out + (size_t)si*64 + c, __float_as_int(v));
            }
        }
    }
}

extern "C" void kernel_launch(void* const* d_in, const int* in_sizes, int n_in,
                              void* d_out, int out_size, void* d_ws, size_t ws_size,
                              hipStream_t stream) {
    const float* xyz   = (const float*)d_in[0];
    const int*   pbidx = (const int*)  d_in[1];
    const float* pfeat = (const float*)d_in[2];
    const int*   pind  = (const int*)  d_in[3];
    const float* w1    = (const float*)d_in[4];
    const float* g1    = (const float*)d_in[5];
    const float* b1    = (const float*)d_in[6];
    const float* w2    = (const float*)d_in[7];
    const float* g2    = (const float*)d_in[8];
    const float* b2    = (const float*)d_in[9];
    float* out = (float*)d_out;

    char* ws = (char*)d_ws;
    int*   table = (int*)ws;                                  // 640000 ints
    float* psum  = (float*)(ws + (size_t)TBL_SZ * 4);         // 64*NBLK floats
    float* psq   = psum + 64*NBLK;                            // 64*NBLK floats
    float* bn    = psq  + 64*NBLK;                            // 256 floats: sc1,sh1,sc2,sh2

    k_init <<<(OUT_SZ + 255)/256, 256, 0, stream>>>(table, out);
    k_table<<<(N_PIL + 255)/256, 256, 0, stream>>>(pind, table);
    k_layer<1><<<NBLK, 256, 0, stream>>>(xyz, pbidx, pfeat, table, w1, w2, bn, psum, psq, out);
    k_bnparam<<<64, 256, 0, stream>>>(psum, psq, bn,       bn + 64,  g1, b1);
    k_layer<2><<<NBLK, 256, 0, stream>>>(xyz, pbidx, pfeat, table, w1, w2, bn, psum, psq, out);
    k_bnparam<<<64, 256, 0, stream>>>(psum, psq, bn + 128, bn + 192, g2, b2);
    k_layer<3><<<NBLK, 256, 0, stream>>>(xyz, pbidx, pfeat, table, w1, w2, bn, psum, psq, out);
}